// Rx_layer_79920751444406
// MI455X (gfx1250) — compile-verified
//
#include <hip/hip_runtime.h>
#include <hip/hip_bf16.h>

// Rx layer on 12 qubits: out = (G ⊗ ... ⊗ G) · state,  G = a·I - i·b·X,
// a = cos(w/2), b = sin(w/2).  Factored as out = G6 · S · G6 with S the
// 64x64 reshape of each 4096-long state vector and G6 = G^{⊗6} (symmetric).
// Two complex 64x64x64 matmuls per state via V_WMMA_F32_16X16X4_F32.

typedef __attribute__((ext_vector_type(2))) float v2f;
typedef __attribute__((ext_vector_type(8))) float v8f;

#define DIM 64
#define NSTATE_ELEMS 4096   // 64*64

__device__ __forceinline__ v8f wmma_f32(v2f a, v2f b, v8f c) {
    // (neg_a, A, neg_b, B, c_mod, C, reuse_a, reuse_b) — f32 WMMA: A/B neg must be 0
    return __builtin_amdgcn_wmma_f32_16x16x4_f32(false, a, false, b, (short)0, c,
                                                 false, false);
}

// A-fragment (16x4 f32) of 64x64 row-major M at row-block ti, k-block kb.
// lane<16: M[ti*16+lane][kb*4+{0,1}] ; lane>=16: M[ti*16+lane-16][kb*4+{2,3}]
__device__ __forceinline__ v2f afrag(const float* __restrict__ M, int ti, int kb, int lane) {
    int row = ti * 16 + (lane & 15);
    int col = kb * 4 + ((lane >> 4) << 1);
    const float* p = M + row * DIM + col;
    v2f r; r.x = p[0]; r.y = p[1];
    return r;
}

// B-fragment (4x16 f32) of 64x64 row-major M at k-block kb, col-block tj.
// VGPR0: rows {kb*4+0 (lanes0-15), kb*4+2 (lanes16-31)}, VGPR1: rows +1
__device__ __forceinline__ v2f bfrag(const float* __restrict__ M, int kb, int tj, int lane) {
    int col = tj * 16 + (lane & 15);
    int row = kb * 4 + ((lane >> 4) << 1);
    v2f r; r.x = M[row * DIM + col]; r.y = M[(row + 1) * DIM + col];
    return r;
}

// Build G6 = G^{⊗6}: G6[i][j] = a^(6-h) * b^h * (-i)^h, h = popcount(i^j).
// g[0..4095] = Re, g[4096..8191] = Im.
__global__ void build_g6_kernel(const float* __restrict__ w, float* __restrict__ g) {
    int idx = blockIdx.x * blockDim.x + threadIdx.x;
    if (idx >= NSTATE_ELEMS) return;
    int i = idx >> 6, j = idx & 63;
    float a = __cosf(w[0] * 0.5f);
    float b = __sinf(w[0] * 0.5f);
    int h = __popc(i ^ j);
    float mag = 1.0f;
    for (int q = 0; q < 6 - h; ++q) mag *= a;   // b may be negative: no powf
    for (int q = 0; q < h; ++q)     mag *= b;
    float re = 0.0f, im = 0.0f;
    switch (h & 3) {
        case 0: re =  mag; break;
        case 1: im = -mag; break;
        case 2: re = -mag; break;
        case 3: im =  mag; break;
    }
    g[idx]                = re;
    g[NSTATE_ELEMS + idx] = im;
}

__global__ __launch_bounds__(256) void rx_apply_kernel(
        const float* __restrict__ sre_g, const float* __restrict__ sim_g,
        const float* __restrict__ g6,    float* __restrict__ out) {
    // 64 KB LDS: S (re+im) reused for T after stage 1, plus G6 (re+im).
    __shared__ __align__(16) float smem[4 * NSTATE_ELEMS];
    float* Sre = smem;                       // state re, later T re
    float* Sim = smem + NSTATE_ELEMS;        // state im, later T im
    float* Gre = smem + 2 * NSTATE_ELEMS;    // G6 re
    float* Gim = smem + 3 * NSTATE_ELEMS;    // G6 im

    const int tid   = threadIdx.x;
    const int lane  = tid & 31;
    const int wave  = tid >> 5;              // 8 waves
    const size_t base = (size_t)blockIdx.x * NSTATE_ELEMS;

    // Cooperative load: 4096 floats per array, 256 threads x 4 float4s each.
    {
        const float4* a0 = (const float4*)(sre_g + base);
        const float4* a1 = (const float4*)(sim_g + base);
        const float4* a2 = (const float4*)(g6);
        const float4* a3 = (const float4*)(g6 + NSTATE_ELEMS);
        float4* b0 = (float4*)Sre; float4* b1 = (float4*)Sim;
        float4* b2 = (float4*)Gre; float4* b3 = (float4*)Gim;
        for (int k = 0; k < 4; ++k) {
            int o = tid + k * 256;           // 1024 float4s per array
            b0[o] = a0[o]; b1[o] = a1[o]; b2[o] = a2[o]; b3[o] = a3[o];
        }
    }
    __syncthreads();

    // ---- Stage 1: T = G6 * S (complex), T held in registers -------------
    v8f Tre[2], Tim[2];
    for (int t = 0; t < 2; ++t) {
        int tile = wave * 2 + t;             // 16 tiles over 8 waves
        int ti = tile >> 2, tj = tile & 3;
        v8f ar = {}; v8f ai = {};
        for (int kb = 0; kb < 16; ++kb) {
            v2f Ar = afrag(Gre, ti, kb, lane);
            v2f Ai = afrag(Gim, ti, kb, lane);
            v2f Br = bfrag(Sre, kb, tj, lane);
            v2f Bi = bfrag(Sim, kb, tj, lane);
            v2f nAi; nAi.x = -Ai.x; nAi.y = -Ai.y;   // f32 WMMA has no A-neg
            ar = wmma_f32(Ar,  Br, ar);   // + G_re*S_re
            ar = wmma_f32(nAi, Bi, ar);   // - G_im*S_im
            ai = wmma_f32(Ar,  Bi, ai);   // + G_re*S_im
            ai = wmma_f32(Ai,  Br, ai);   // + G_im*S_re
        }
        Tre[t] = ar; Tim[t] = ai;
    }
    __syncthreads();                          // all reads of S complete

    // Write T into S's LDS slot (D layout: VGPR r -> row r + 8*(lane>=16)).
    for (int t = 0; t < 2; ++t) {
        int tile = wave * 2 + t;
        int ti = tile >> 2, tj = tile & 3;
        int col   = tj * 16 + (lane & 15);
        int rbase = ti * 16 + ((lane >> 4) << 3);
        for (int r = 0; r < 8; ++r) {
            Sre[(rbase + r) * DIM + col] = Tre[t][r];
            Sim[(rbase + r) * DIM + col] = Tim[t][r];
        }
    }
    __syncthreads();

    // ---- Stage 2: O = T * G6 (complex), store interleaved complex64 -----
    for (int t = 0; t < 2; ++t) {
        int tile = wave * 2 + t;
        int ti = tile >> 2, tj = tile & 3;
        v8f ore = {}; v8f oim = {};
        for (int kb = 0; kb < 16; ++kb) {
            v2f Ar = afrag(Sre, ti, kb, lane);   // T_re
            v2f Ai = afrag(Sim, ti, kb, lane);   // T_im
            v2f Br = bfrag(Gre, kb, tj, lane);
            v2f Bi = bfrag(Gim, kb, tj, lane);
            v2f nAi; nAi.x = -Ai.x; nAi.y = -Ai.y;
            ore = wmma_f32(Ar,  Br, ore);
            ore = wmma_f32(nAi, Bi, ore);
            oim = wmma_f32(Ar,  Bi, oim);
            oim = wmma_f32(Ai,  Br, oim);
        }
        int col   = tj * 16 + (lane & 15);
        int rbase = ti * 16 + ((lane >> 4) << 3);
        float2* out2 = (float2*)out;             // complex64 = interleaved (re, im)
        for (int r = 0; r < 8; ++r) {
            int idx = (rbase + r) * DIM + col;
            float2 v; v.x = ore[r]; v.y = oim[r];
            out2[base + idx] = v;                 // 8-byte coalesced store
        }
    }
}

extern "C" void kernel_launch(void* const* d_in, const int* in_sizes, int n_in,
                              void* d_out, int out_size, void* d_ws, size_t ws_size,
                              hipStream_t stream) {
    const float* state_re = (const float*)d_in[0];   // (256,1,2,4096,1) f32
    const float* state_im = (const float*)d_in[1];   // (256,1,2,4096,1) f32
    const float* weights  = (const float*)d_in[2];   // (1,) f32
    // d_in[3] = n_qubits (fixed 12 by setup)

    float* g6 = (float*)d_ws;                        // 8192 floats = 32 KB

    int nstates = in_sizes[0] / NSTATE_ELEMS;        // 512

    build_g6_kernel<<<(NSTATE_ELEMS + 255) / 256, 256, 0, stream>>>(weights, g6);
    rx_apply_kernel<<<nstates, 256, 0, stream>>>(state_re, state_im, g6,
                                                 (float*)d_out);
}